// AmpChi_90125593739870
// MI455X (gfx1250) — compile-verified
//
#include <hip/hip_runtime.h>

#define LXn 10
#define LYn 10
#define Dn 4
#define PHYSn 2
#define CHIn 4
#define Bn 4096

typedef float v2f __attribute__((ext_vector_type(2)));
typedef float v8f __attribute__((ext_vector_type(8)));
typedef unsigned int u32;
typedef u32 u32x4 __attribute__((ext_vector_type(4)));
typedef u32 u32x8 __attribute__((ext_vector_type(8)));

__device__ inline void wg_sync() { __syncthreads(); }

__device__ inline int ctzi(int v) { return (int)__builtin_ctz((unsigned)v); }

__device__ inline void zmem(float* p, int n) {
  for (int i = (int)threadIdx.x; i < n; i += 32) p[i] = 0.f;
}

// C(16x16) = A(16xK) * B(Kx16), arbitrary element strides, K multiple of 4.
// Layout per CDNA5 ISA: A 16x4 f32 -> 2 VGPRs (lanes 0-15: K=0/1, lanes 16-31: K=2/3),
// C/D 16x16 f32 -> 8 VGPRs (VGPR r, lane l: M = r + 8*(l>>4), N = l&15).
__device__ inline void wmma_mm(const float* A, int sam, int sak,
                               const float* Bm, int sbk, int sbn,
                               float* C, int scm, int scn, int K)
{
  const int lane = (int)threadIdx.x & 31;
  const int mr = lane & 15;
  const int hi = lane >> 4;
  v8f acc = {0.f, 0.f, 0.f, 0.f, 0.f, 0.f, 0.f, 0.f};
  for (int k0 = 0; k0 < K; k0 += 4) {
    const int ka = k0 + 2 * hi;
    v2f a, b;
    a.x = A[mr * sam + (ka + 0) * sak];
    a.y = A[mr * sam + (ka + 1) * sak];
    b.x = Bm[(ka + 0) * sbk + mr * sbn];
    b.y = Bm[(ka + 1) * sbk + mr * sbn];
    acc = __builtin_amdgcn_wmma_f32_16x16x4_f32(false, a, false, b,
                                                (short)0, acc, false, false);
  }
#pragma unroll
  for (int r = 0; r < 8; ++r)
    C[(r + 8 * hi) * scm + mr * scn] = acc[r];
}

// Lane-parallel two-sided Jacobi eigensolver on padded symmetric 16x16 G
// (PSD Gram matrices; rows/cols >= n are zero -> identity rotations there).
// Round-robin tournament: 8 disjoint pairs/round, 15 rounds/sweep. Rotations
// on disjoint 2-D subspaces commute, so simultaneous row-phase then col-phase
// application is exact. Eigenvectors in V columns, eigenvalues ev[] desc.
__device__ void jacobi_eig_par(float* G, float* V, float* ev, float* cs)
{
  const int lane = (int)threadIdx.x;
  for (int idx = lane; idx < 256; idx += 32)
    V[idx] = ((idx >> 4) == (idx & 15)) ? 1.f : 0.f;
  wg_sync();
  for (int sweep = 0; sweep < 8; ++sweep) {
    for (int rr = 0; rr < 15; ++rr) {
      // rotation params: one pair per lane (pairs disjoint, cover all 16)
      if (lane < 8) {
        int a = (rr + lane) % 15;
        int b = (lane == 0) ? 15 : (rr + 15 - lane) % 15;
        int p = a < b ? a : b;
        int q = a < b ? b : a;
        float gpq = G[p * 16 + q];
        float c = 1.f, s = 0.f;
        if (fabsf(gpq) > 1e-14f) {
          float gpp = G[p * 16 + p], gqq = G[q * 16 + q];
          float tau = (gqq - gpp) / (2.f * gpq);
          float t = ((tau >= 0.f) ? 1.f : -1.f) /
                    (fabsf(tau) + sqrtf(1.f + tau * tau));
          c = 1.f / sqrtf(1.f + t * t);
          s = t * c;
        }
        cs[lane * 2] = c;
        cs[lane * 2 + 1] = s;
      }
      wg_sync();
      // row phase: J^T G  (8 pairs x 16 cols, disjoint rows)
      for (int item = lane; item < 128; item += 32) {
        int i = item >> 4, col = item & 15;
        int a = (rr + i) % 15;
        int b = (i == 0) ? 15 : (rr + 15 - i) % 15;
        int p = a < b ? a : b;
        int q = a < b ? b : a;
        float c = cs[i * 2], s = cs[i * 2 + 1];
        float gp = G[p * 16 + col], gq = G[q * 16 + col];
        G[p * 16 + col] = c * gp - s * gq;
        G[q * 16 + col] = s * gp + c * gq;
      }
      wg_sync();
      // col phase: (J^T G) J on G, and V J  (disjoint cols)
      for (int item = lane; item < 128; item += 32) {
        int i = item >> 4, row = item & 15;
        int a = (rr + i) % 15;
        int b = (i == 0) ? 15 : (rr + 15 - i) % 15;
        int p = a < b ? a : b;
        int q = a < b ? b : a;
        float c = cs[i * 2], s = cs[i * 2 + 1];
        float gp = G[row * 16 + p], gq = G[row * 16 + q];
        G[row * 16 + p] = c * gp - s * gq;
        G[row * 16 + q] = s * gp + c * gq;
        float vp = V[row * 16 + p], vq = V[row * 16 + q];
        V[row * 16 + p] = c * vp - s * vq;
        V[row * 16 + q] = s * vp + c * vq;
      }
      wg_sync();
    }
  }
  // eigenvalues + descending selection sort (tiny; lane 0)
  if (lane == 0) {
    for (int i = 0; i < 16; ++i) ev[i] = G[i * 16 + i];
    for (int i = 0; i < 16; ++i) {
      int bi = i;
      for (int j = i + 1; j < 16; ++j)
        if (ev[j] > ev[bi]) bi = j;
      if (bi != i) {
        float t = ev[i]; ev[i] = ev[bi]; ev[bi] = t;
        for (int kk = 0; kk < 16; ++kk) {
          float t2 = V[kk * 16 + i];
          V[kk * 16 + i] = V[kk * 16 + bi];
          V[kk * 16 + bi] = t2;
        }
      }
    }
  }
  wg_sync();
}

// Stage one peps row (10 sites * 256 * PHYS = 5120 f32 = 20KB, contiguous)
// into LDS with the Tensor Data Mover. 1-D tile: data_size=4B, dim0=5120.
__device__ inline void tdm_load_row(const float* peps, int row, float* stage)
{
  u32 lds_off = (u32)(uintptr_t)stage;   // flat LDS addr: low 32 bits = LDS offset
  unsigned long long ga = (unsigned long long)(uintptr_t)(peps + (size_t)row * 5120);
  lds_off = (u32)__builtin_amdgcn_readfirstlane((int)lds_off);
  u32 ga_lo = (u32)__builtin_amdgcn_readfirstlane((int)(u32)ga);
  u32 ga_hi = (u32)__builtin_amdgcn_readfirstlane((int)(u32)(ga >> 32));
  u32x4 g0;
  g0[0] = 1u;                                  // count=1, user descriptor
  g0[1] = lds_off;                             // lds_addr (bytes)
  g0[2] = ga_lo;                               // global_addr[31:0]
  g0[3] = (ga_hi & 0x01FFFFFFu) | (2u << 30);  // global_addr[56:32] | type=2
  u32x8 g1;
  g1[0] = 2u << 16;      // workgroup_mask=0, data_size=2 (4 bytes)
  g1[1] = 5120u << 16;   // tensor_dim0[15:0] in bits[63:48]
  g1[2] = 1u << 16;      // tensor_dim0 hi=0 | tensor_dim1=1 (low half)
  g1[3] = 5120u << 16;   // tensor_dim1 hi=0 | tile_dim0=5120
  g1[4] = 1u;            // tile_dim1=1, tile_dim2=0
  g1[5] = 5120u;         // tensor_dim0_stride low
  g1[6] = 0u;
  g1[7] = 0u;
  asm volatile("tensor_load_to_lds %0, %1" :: "s"(g0), "s"(g1) : "memory");
  __builtin_amdgcn_s_wait_tensorcnt(0);
}

struct Ws {
  float *S1, *S2, *S3, *G, *V, *Bs, *SV, *ev, *red, *Ea, *Eb;
  int *bld, *brd, *bldS, *brdS;
};

// All bond dims are powers of two throughout: init {1,4}; absorption scales by
// {1,4}; right sweep k=min(l,4r); left sweep k=min(4,4l,r) with l,r in {1,4,16}.
// So every div/mod below is a ctz-shift/mask.

// Right-canonize (Gram-eigen == SVD subspace) then left-to-right truncate to chi,
// then normalize tensors accumulating log-norm. Matches _canonize_compress.
__device__ float compress_row(float* T, Ws w)
{
  const int lane = (int)threadIdx.x;

  // ---- right-to-left sweep (full SVD, k = min(l, r*d)) ----
  for (int j = LYn - 1; j >= 1; --j) {
    int l = w.bld[j];
    int rd = w.brd[j] * 4;
    float* Mj = T + j * 1024;
    // G = M M^T  (l x l), K = rd
    wmma_mm(Mj, 64, 1, Mj, 1, 64, w.G, 16, 1, rd);
    wg_sync();
    jacobi_eig_par(w.G, w.V, w.ev, w.red);
    int k = (l < rd) ? l : rd;
    int Kl = (l + 3) & ~3;
    // W = U^T M  (k x rd), U = V columns
    for (int t4 = 0; t4 < 4; ++t4)
      wmma_mm(w.V, 1, 16, Mj + 16 * t4, 64, 1, w.S2 + 16 * t4, 64, 1, Kl);
    wg_sync();
    // Vt = diag(1/s) W ; zero discarded rows
    for (int idx = lane; idx < 1024; idx += 32) {
      int m = idx >> 6;
      float is = 0.f;
      if (m < k) {
        float sv = sqrtf(fmaxf(w.ev[m], 0.f));
        if (sv > 1e-20f) is = 1.f / sv;
      }
      w.S2[idx] *= is;
    }
    // US[b][m] = U[b][m] * s[m]
    for (int idx = lane; idx < 256; idx += 32) {
      int b = idx >> 4, m2 = idx & 15;
      float val = 0.f;
      if (b < l && m2 < k)
        val = w.V[idx] * sqrtf(fmaxf(w.ev[m2], 0.f));
      w.SV[idx] = val;
    }
    wg_sync();
    // absorb US into mps[j-1]: out[a,m,d] = sum_b prev[a,b,d] * US[b,m]
    float* Pj = T + (j - 1) * 1024;
    for (int d = 0; d < 4; ++d)
      wmma_mm(Pj + d, 64, 4, w.SV, 16, 1, w.S3 + d, 64, 4, Kl);
    wg_sync();
    for (int idx = lane; idx < 1024; idx += 32) {
      Mj[idx] = w.S2[idx];
      Pj[idx] = w.S3[idx];
    }
    wg_sync();
    if (lane == 0) { w.bld[j] = k; w.brd[j - 1] = k; }
    wg_sync();
  }

  // ---- left-to-right truncating sweep (k = min(chi, l*d, r)) ----
  for (int j = 0; j < LYn - 1; ++j) {
    int l = w.bld[j], r = w.brd[j];
    int ld = l * 4;
    int rs = ctzi(r);
    float* Mj = T + j * 1024;
    // m = transpose(mps[j],(0,2,1)).reshape(l*d, r) into S1 (64x16)
    zmem(w.S1, 1024);
    wg_sync();
    for (int idx = lane; idx < ld * r; idx += 32) {
      int ri = idx & (r - 1);
      int t1 = idx >> rs;  // li*4 + d
      w.S1[t1 * 16 + ri] = Mj[(t1 >> 2) * 64 + ri * 4 + (t1 & 3)];
    }
    wg_sync();
    // G = m^T m (r x r), K = ld
    wmma_mm(w.S1, 1, 16, w.S1, 16, 1, w.G, 16, 1, ld);
    wg_sync();
    jacobi_eig_par(w.G, w.V, w.ev, w.red);
    int k = CHIn; if (ld < k) k = ld; if (r < k) k = r;
    int ks = ctzi(k);
    int Kr = (r + 3) & ~3;
    // U2 = m * V (64 x 16)
    for (int t4 = 0; t4 < 4; ++t4)
      wmma_mm(w.S1 + 256 * t4, 16, 1, w.V, 16, 1, w.S2 + 256 * t4, 16, 1, Kr);
    wg_sync();
    for (int idx = lane; idx < 1024; idx += 32) {
      int m2 = idx & 15;
      float is = 0.f;
      if (m2 < k) {
        float sv = sqrtf(fmaxf(w.ev[m2], 0.f));
        if (sv > 1e-20f) is = 1.f / sv;
      }
      w.S2[idx] *= is;
    }
    wg_sync();
    zmem(Mj, 1024);
    wg_sync();
    // mps[j] = transpose(u.reshape(l,d,k),(0,2,1))
    for (int idx = lane; idx < ld * k; idx += 32) {
      int kk = idx & (k - 1);
      int t1 = idx >> ks;  // li*4 + d
      Mj[(t1 >> 2) * 64 + kk * 4 + (t1 & 3)] = w.S2[t1 * 16 + kk];
    }
    // SV = diag(s) V^T  (k x r)
    for (int idx = lane; idx < 256; idx += 32) {
      int kk = idx >> 4, b = idx & 15;
      float val = 0.f;
      if (kk < k && b < r)
        val = sqrtf(fmaxf(w.ev[kk], 0.f)) * w.V[b * 16 + kk];
      w.SV[idx] = val;
    }
    wg_sync();
    // absorb into mps[j+1]: out[m,rr,d] = sum_b SV[m,b] * next[b,rr,d]
    float* Nj = T + (j + 1) * 1024;
    for (int d = 0; d < 4; ++d)
      wmma_mm(w.SV, 16, 1, Nj + d, 64, 4, w.S3 + d, 64, 4, Kr);
    wg_sync();
    for (int idx = lane; idx < 1024; idx += 32) Nj[idx] = w.S3[idx];
    wg_sync();
    if (lane == 0) { w.brd[j] = k; w.bld[j + 1] = k; }
    wg_sync();
  }

  // ---- strip norms ----
  float logn = 0.f;
  for (int j = 0; j < LYn; ++j) {
    float* Mj = T + j * 1024;
    float p = 0.f;
    for (int idx = lane; idx < 1024; idx += 32) { float v = Mj[idx]; p += v * v; }
    w.red[lane] = p;
    wg_sync();
    if (lane == 0) {
      float ss = 0.f;
      for (int q = 0; q < 32; ++q) ss += w.red[q];
      w.red[0] = ss;
    }
    wg_sync();
    float nrm = sqrtf(w.red[0]);
    float inv = (nrm > 0.f) ? 1.f / nrm : 0.f;
    for (int idx = lane; idx < 1024; idx += 32) Mj[idx] *= inv;
    logn += logf(fmaxf(nrm, 1e-38f));
    wg_sync();
  }
  return logn;
}

// One row absorption:
//   bottom: 'arv,vwlc->alrcw'   top: 'arw,vwlc->alrcv'
// done as  P[(a,r),(x,l,c)] = M[(a,r),K] * B[K,(x,l,c)]  with one WMMA row of tiles,
// then an index-scatter into the padded MPS layout.
__device__ void absorb_row(const float* cur, float* nxt, const float* stage,
                           const int* x, int samp, int row, int isTop, Ws w)
{
  const int lane = (int)threadIdx.x;
  for (int j = 0; j < LYn; ++j) {
    int l2 = (j == 0) ? 1 : 4;
    int c2 = (j == LYn - 1) ? 1 : 4;
    int l2s = ctzi(l2), c2s = ctzi(c2);
    int ba = w.bld[j], br_ = w.brd[j];
    int brs = ctzi(br_);
    int phys = x[samp * (LXn * LYn) + row * LYn + j];
    zmem(w.Bs, 256);
    zmem(w.S1, 64);
    wg_sync();
    // B matrix (contraction dim x (other, l, c))
    for (int idx = lane; idx < 16 * l2 * c2; idx += 32) {
      int c = idx & (c2 - 1); int t1 = idx >> c2s;
      int li = t1 & (l2 - 1); int t2 = t1 >> l2s;
      int wq = t2 & 3;        int vq = t2 >> 2;
      float val = stage[j * 512 + ((((vq * 4 + wq) * 4 + li) * 4 + c) * 2) + phys];
      if (!isTop)
        w.Bs[vq * 64 + (wq * l2 + li) * c2 + c] = val;  // contract over v
      else
        w.Bs[wq * 64 + (vq * l2 + li) * c2 + c] = val;  // contract over w
    }
    // M matrix ((a,r) x 4)
    for (int idx = lane; idx < ba * br_ * 4; idx += 32) {
      int vv = idx & 3; int t1 = idx >> 2;
      int r = t1 & (br_ - 1); int a = t1 >> brs;
      w.S1[(a * br_ + r) * 4 + vv] = cur[j * 1024 + a * 64 + r * 4 + vv];
    }
    wg_sync();
    for (int t4 = 0; t4 < 4; ++t4)
      wmma_mm(w.S1, 4, 1, w.Bs + 16 * t4, 64, 1, w.S2 + 16 * t4, 64, 1, 4);
    wg_sync();
    float* nj = nxt + j * 1024;
    zmem(nj, 1024);
    wg_sync();
    int tot = ba * br_ * l2 * c2 * 4;
    for (int idx = lane; idx < tot; idx += 32) {
      int wq = idx & 3;       int t1 = idx >> 2;
      int c = t1 & (c2 - 1);  t1 >>= c2s;
      int li = t1 & (l2 - 1); t1 >>= l2s;
      int r = t1 & (br_ - 1); int a = t1 >> brs;
      nj[(a * l2 + li) * 64 + (r * c2 + c) * 4 + wq] =
          w.S2[(a * br_ + r) * 64 + (wq * l2 + li) * c2 + c];
    }
    wg_sync();
  }
  if (lane == 0) {
    for (int j = 0; j < LYn; ++j) {
      w.bld[j] *= (j == 0) ? 1 : 4;
      w.brd[j] *= (j == LYn - 1) ? 1 : 4;
    }
  }
  wg_sync();
}

__device__ void init_boundary(float* T, const float* stage, const int* x,
                              int samp, int row, int isTop, Ws w)
{
  const int lane = (int)threadIdx.x;
  zmem(T, 10240);
  wg_sync();
  for (int j = 0; j < LYn; ++j) {
    int l2 = (j == 0) ? 1 : 4;
    int c2 = (j == LYn - 1) ? 1 : 4;
    int c2s = ctzi(c2);
    int phys = x[samp * (LXn * LYn) + row * LYn + j];
    for (int idx = lane; idx < l2 * c2 * 4; idx += 32) {
      int v3 = idx & 3; int t1 = idx >> 2;
      int c = t1 & (c2 - 1); int li = t1 >> c2s;
      int si;
      if (!isTop)  // bottom row: t[0, w, l, c] -> (l, c, w)
        si = (((0 * 4 + v3) * 4 + li) * 4 + c) * 2 + phys;
      else         // top row:    t[v, 0, l, c] -> (l, c, v)
        si = (((v3 * 4 + 0) * 4 + li) * 4 + c) * 2 + phys;
      T[j * 1024 + li * 64 + c * 4 + v3] = stage[j * 512 + si];
    }
  }
  if (lane == 0) {
    for (int j = 0; j < LYn; ++j) {
      w.bld[j] = (j == 0) ? 1 : 4;
      w.brd[j] = (j == LYn - 1) ? 1 : 4;
    }
  }
  wg_sync();
}

// LDS budget (floats): 3*10240 MPS buffers + 5120 stage + 3*1024 scratch
// + 4*256 small mats + 80 misc + 64 (ints) = 40080 -> 160320 bytes.
#define LDS_FLOATS 40080
#define LDS_BYTES  (LDS_FLOATS * 4)

extern "C" __global__ __launch_bounds__(32)
void peps_amp_kernel(const int* __restrict__ x, const float* __restrict__ peps,
                     float* __restrict__ out)
{
  extern __shared__ float lds[];
  const int lane = (int)threadIdx.x;
  const int samp = (int)blockIdx.x;

  float* bufA = lds;
  float* bufB = bufA + 10240;
  float* bufC = bufB + 10240;
  float* stage = bufC + 10240;
  float* S1 = stage + 5120;
  float* S2 = S1 + 1024;
  float* S3 = S2 + 1024;
  float* G  = S3 + 1024;
  float* V  = G + 256;
  float* Bs = V + 256;
  float* SV = Bs + 256;
  float* ev = SV + 256;
  float* red = ev + 16;
  float* Ea  = red + 32;
  float* Eb  = Ea + 16;
  int* bld  = (int*)(Eb + 16);
  int* brd  = bld + 16;
  int* bldS = brd + 16;
  int* brdS = bldS + 16;

  Ws w{S1, S2, S3, G, V, Bs, SV, ev, red, Ea, Eb, bld, brd, bldS, brdS};

  float logn = 0.f;

  // ---- bottom boundary MPS ----
  tdm_load_row(peps, 0, stage);
  wg_sync();
  init_boundary(bufA, stage, x, samp, 0, 0, w);

  float* cur = bufA;
  float* nxt = bufB;
  for (int i = 1; i < LXn / 2; ++i) {
    tdm_load_row(peps, i, stage);
    wg_sync();
    absorb_row(cur, nxt, stage, x, samp, i, 0, w);
    logn += compress_row(nxt, w);
    float* t = cur; cur = nxt; nxt = t;
  }
  float* bot = cur;  // 4 iterations -> ends in bufA
  if (lane == 0)
    for (int j = 0; j < LYn; ++j) { bldS[j] = bld[j]; brdS[j] = brd[j]; }
  wg_sync();

  // ---- top boundary MPS ----
  tdm_load_row(peps, LXn - 1, stage);
  wg_sync();
  init_boundary(bufB, stage, x, samp, LXn - 1, 1, w);
  cur = bufB; nxt = bufC;
  for (int i = LXn - 2; i >= LXn / 2; --i) {
    tdm_load_row(peps, i, stage);
    wg_sync();
    absorb_row(cur, nxt, stage, x, samp, i, 1, w);
    logn += compress_row(nxt, w);
    float* t = cur; cur = nxt; nxt = t;
  }
  float* top = cur;  // 4 iterations -> ends in bufB

  // ---- final zip: E <- 'ab,acd,bed->ce' ----
  if (lane == 0) {
    Ea[0] = 1.f;
    int na = 1, nb = 1;
    for (int j = 0; j < LYn; ++j) {
      int nc = brdS[j];
      int ne = brd[j];
      for (int c = 0; c < nc; ++c)
        for (int e = 0; e < ne; ++e) {
          float acc = 0.f;
          for (int a = 0; a < na; ++a)
            for (int b = 0; b < nb; ++b) {
              float eab = Ea[a * 4 + b];
              for (int d = 0; d < 4; ++d)
                acc += eab * bot[j * 1024 + a * 64 + c * 4 + d]
                           * top[j * 1024 + b * 64 + e * 4 + d];
            }
          Eb[c * 4 + e] = acc;
        }
      na = nc; nb = ne;
      for (int q = 0; q < 16; ++q) Ea[q] = Eb[q];
    }
    out[samp] = Ea[0] * expf(logn);
  }
}

extern "C" void kernel_launch(void* const* d_in, const int* in_sizes, int n_in,
                              void* d_out, int out_size, void* d_ws, size_t ws_size,
                              hipStream_t stream)
{
  (void)in_sizes; (void)n_in; (void)d_ws; (void)ws_size; (void)out_size;
  const int* x = (const int*)d_in[0];
  const float* peps = (const float*)d_in[1];
  float* out = (float*)d_out;
  peps_amp_kernel<<<dim3(Bn), dim3(32), LDS_BYTES, stream>>>(x, peps, out);
}